// MinimalQuantumLayer_7851200217707
// MI455X (gfx1250) — compile-verified
//
#include <hip/hip_runtime.h>
#include <hip/hip_bf16.h>

typedef __attribute__((ext_vector_type(2))) float v2f;
typedef __attribute__((ext_vector_type(8))) float v8f;

// ------------------------------------------------------------------
// Kernel 1: evolve the 16 basis states through the fixed circuit and
// store W[n][m] = U[m][n] (real part then imag part) into workspace.
// Thread j owns column j of U (16 complex amplitudes in registers).
// ------------------------------------------------------------------
__global__ __launch_bounds__(32) void build_unitary(const float* __restrict__ qw,
                                                    float* __restrict__ ws) {
  const int j = threadIdx.x;
  if (j >= 16) return;

  float re[16], im[16];
#pragma unroll
  for (int n = 0; n < 16; ++n) { re[n] = (n == j) ? 1.0f : 0.0f; im[n] = 0.0f; }

#pragma unroll
  for (int l = 0; l < 2; ++l) {
    // RX(w) on each qubit q (qubit 0 == bit 3 of the flat index)
#pragma unroll
    for (int q = 0; q < 4; ++q) {
      const float w = qw[l * 4 + q];
      const float c = __cosf(0.5f * w);
      const float s = __sinf(0.5f * w);
      const int bit = 1 << (3 - q);
#pragma unroll
      for (int n0 = 0; n0 < 16; ++n0) {
        if (n0 & bit) continue;
        const int n1 = n0 | bit;
        const float r0 = re[n0], i0 = im[n0], r1 = re[n1], i1 = im[n1];
        // [[c, -i s], [-i s, c]]
        re[n0] = c * r0 + s * i1;
        im[n0] = c * i0 - s * r1;
        re[n1] = s * i0 + c * r1;
        im[n1] = c * i1 - s * r0;
      }
    }
    // CNOT ring: control q -> target (q+1)%4
#pragma unroll
    for (int q = 0; q < 4; ++q) {
      const int bc = 1 << (3 - q);
      const int bt = 1 << (3 - ((q + 1) & 3));
#pragma unroll
      for (int n = 0; n < 16; ++n) {
        if ((n & bc) && !(n & bt)) {
          const int n2 = n | bt;
          float tr = re[n]; re[n] = re[n2]; re[n2] = tr;
          float ti = im[n]; im[n] = im[n2]; im[n2] = ti;
        }
      }
    }
  }

#pragma unroll
  for (int m = 0; m < 16; ++m) {
    ws[j * 16 + m]       = re[m];   // W_re[n=j][m] = Re U[m][j]
    ws[256 + j * 16 + m] = im[m];   // W_im[n=j][m] = Im U[m][j]
  }
}

// ------------------------------------------------------------------
// Kernel 2: one wave handles 16 consecutive patches per iteration.
// Transposed GEMM:  D[m][patch] = U * Psi^T
//   A (constant, 16x16 via 4 chunks of 16x4): Re/Im U[m][n]
//   B (per patch,  4 chunks of 4x16):         Psi[n][patch]
// D layout => lane owns ONE patch (col) and all 16 m along VGPRs
// (m = r + 8*hi), so the signed <Z_q> sums are an in-register 8-pt
// WHT + one shfl_xor(16) per qubit, and the store is one b128/lane.
// ------------------------------------------------------------------
__global__ __launch_bounds__(256) void mql_main(const float* __restrict__ x,
                                                const float* __restrict__ ws,
                                                float* __restrict__ out,
                                                int numTiles) {
  const int lane = threadIdx.x & 31;
  const int col  = lane & 15;      // patch within tile (B cols / D cols)
  const int hi   = lane >> 4;      // half-wave select
  const int waveId   = blockIdx.x * (blockDim.x >> 5) + (threadIdx.x >> 5);
  const int numWaves = gridDim.x * (blockDim.x >> 5);

  const float* __restrict__ Wre = ws;
  const float* __restrict__ Wim = ws + 256;

  // Constant A operand: chunk kc, VGPR j holds A[M=col][K = 4*kc+2*hi+j]
  //   A_re[m][n] = Re U[m][n] = ws[n*16 + m]
  v2f are[4], aim[4];
#pragma unroll
  for (int kc = 0; kc < 4; ++kc) {
    const int n0 = 4 * kc + 2 * hi;
    are[kc].x = Wre[n0 * 16 + col];
    are[kc].y = Wre[(n0 + 1) * 16 + col];
    aim[kc].x = Wim[n0 * 16 + col];
    aim[kc].y = Wim[(n0 + 1) * 16 + col];
  }

  const float QPI = 0.78539816339744830962f;  // pi/4
  float4* __restrict__ out4 = (float4*)out;

  for (int t = waveId; t < numTiles; t += numWaves) {
    // ---- load the 4 angles of this lane's patch (coalesced float2) ----
    const int P   = t * 16 + col;
    const int b   = P >> 14;          // / (128*128)
    const int rem = P & 16383;
    const int oh  = rem >> 7;
    const int ow  = rem & 127;
    const long long idx = (((long long)b * 256 + 2 * oh) * 256) + 2 * ow;
    const float2* xp = (const float2*)(x + idx);
    const float2 r0 = xp[0];      // x[2oh][2ow], x[2oh][2ow+1]
    const float2 r1 = xp[128];    // x[2oh+1][2ow], x[2oh+1][2ow+1]

    float s0, c0, s1, c1, s2, c2, s3, c3;
    __sincosf(r0.x * QPI, &s0, &c0);
    __sincosf(r0.y * QPI, &s1, &c1);
    __sincosf(r1.x * QPI, &s2, &c2);
    __sincosf(r1.y * QPI, &s3, &c3);

    // Psi[n] = (n&8?s0:c0)(n&4?s1:c1)(n&2?s2:c2)(n&1?s3:c3)
    const float p01[4] = {c0 * c1, c0 * s1, s0 * c1, s0 * s1};  // index n>>2
    const float e2 = hi ? s2 : c2;       // this half-wave's n&2 selector
    const float w0 = e2 * c3;            // n parity j = 0
    const float w1 = e2 * s3;            // n parity j = 1

    v8f accre = {0.f, 0.f, 0.f, 0.f, 0.f, 0.f, 0.f, 0.f};
    v8f accim = {0.f, 0.f, 0.f, 0.f, 0.f, 0.f, 0.f, 0.f};

#pragma unroll
    for (int kc = 0; kc < 4; ++kc) {
      // B operand: VGPR j holds Psi[n = 4*kc + 2*hi + j][patch = col]
      v2f Bm;
      Bm.x = p01[kc] * w0;
      Bm.y = p01[kc] * w1;
      accre = __builtin_amdgcn_wmma_f32_16x16x4_f32(
          false, are[kc], false, Bm, (short)0, accre, false, false);
      accim = __builtin_amdgcn_wmma_f32_16x16x4_f32(
          false, aim[kc], false, Bm, (short)0, accim, false, false);
    }

    // ---- probabilities of the 8 basis states this lane holds ----
    float p[8];
#pragma unroll
    for (int r = 0; r < 8; ++r) {
      p[r] = accre[r] * accre[r] + accim[r] * accim[r];
    }

    // ---- in-register 8-point signed sums over m-bits (m = r + 8*hi) ----
    const float t0 = p[0] + p[1], t1 = p[2] + p[3];
    const float t2 = p[4] + p[5], t3 = p[6] + p[7];
    const float u0 = p[0] - p[1], u1 = p[2] - p[3];
    const float u2 = p[4] - p[5], u3 = p[6] - p[7];
    const float v01 = t0 + t1, v23 = t2 + t3;

    float o3 = (u0 + u1) + (u2 + u3);          // sign from m bit0
    float o2 = (t0 - t1) + (t2 - t3);          // sign from m bit1
    float o1 = v01 - v23;                      // sign from m bit2
    float o0 = hi ? -(v01 + v23) : (v01 + v23);// sign from m bit3 (= hi)

    // ---- combine the two half-waves (partner holds m ^ 8) ----
    o0 += __shfl_xor(o0, 16, 32);
    o1 += __shfl_xor(o1, 16, 32);
    o2 += __shfl_xor(o2, 16, 32);
    o3 += __shfl_xor(o3, 16, 32);

    // ---- one coalesced 16B store per lane (low half only) ----
    if (hi == 0) {
      out4[P] = make_float4(o0, o1, o2, o3);
    }
  }
}

extern "C" void kernel_launch(void* const* d_in, const int* in_sizes, int n_in,
                              void* d_out, int out_size, void* d_ws, size_t ws_size,
                              hipStream_t stream) {
  const float* x  = (const float*)d_in[0];   // (32,256,256,1) f32
  const float* qw = (const float*)d_in[1];   // (2,4) f32
  float* out = (float*)d_out;                // (32,128,128,4) f32
  float* ws  = (float*)d_ws;                 // needs 2 KB for U

  const int numTiles = in_sizes[0] / 64;     // (#patches)/16 = elems/4/16

  build_unitary<<<1, 32, 0, stream>>>(qw, ws);
  mql_main<<<1024, 256, 0, stream>>>(x, ws, out, numTiles);
}